// GAT_4423816315316
// MI455X (gfx1250) — compile-verified
//
#include <hip/hip_runtime.h>
#include <hip/hip_bf16.h>

typedef __attribute__((ext_vector_type(2))) float v2f;
typedef __attribute__((ext_vector_type(8))) float v8f;

#define NEG_SLOPE 0.2f

// ---------------------------------------------------------------------------
// WMMA GEMM: C[M,NOUT] = A[M,K] * B[K,NOUT], row-major, f32 via
// V_WMMA_F32_16X16X4_F32. One wave per 16-row tile; NOUT/16 accumulators.
// ---------------------------------------------------------------------------
template <int K, int NOUT>
__global__ void wmma_gemm_f32(const float* __restrict__ A,
                              const float* __restrict__ B,
                              float* __restrict__ C, int M) {
  const int lane = threadIdx.x & 31;
  const int wave = threadIdx.x >> 5;
  const int wavesPerBlock = blockDim.x >> 5;
  const int tile = blockIdx.x * wavesPerBlock + wave;  // wave-uniform
  const int mtiles = (M + 15) >> 4;
  if (tile >= mtiles) return;  // uniform exit: EXEC all-1 for WMMA below

  const int row0 = tile * 16;
  const int m = lane & 15;
  const int kb = (lane >> 4) * 2;  // lanes 0-15: K pair {k,k+1}; 16-31: {k+2,k+3}

  constexpr int NT = NOUT / 16;
  v8f acc[NT];
#pragma unroll
  for (int t = 0; t < NT; ++t) acc[t] = (v8f){0.f, 0.f, 0.f, 0.f, 0.f, 0.f, 0.f, 0.f};

  const float* __restrict__ arow = A + (size_t)(row0 + m) * K;
  const int n0 = lane & 15;

  for (int k = 0; k < K; k += 4) {
    v2f a;
    a.x = arow[k + kb];
    a.y = arow[k + kb + 1];
#pragma unroll
    for (int t = 0; t < NT; ++t) {
      const int n = t * 16 + n0;
      v2f b;
      b.x = B[(size_t)(k + kb) * NOUT + n];
      b.y = B[(size_t)(k + kb + 1) * NOUT + n];
      acc[t] = __builtin_amdgcn_wmma_f32_16x16x4_f32(
          /*neg_a=*/false, a, /*neg_b=*/false, b,
          /*c_mod=*/(short)0, acc[t], /*reuse_a=*/false, /*reuse_b=*/false);
    }
  }

  // C/D layout: VGPR r -> row (r + (lane>=16 ? 8 : 0)), col = lane&15
  const int rowoff = (lane >> 4) * 8;
#pragma unroll
  for (int t = 0; t < NT; ++t) {
#pragma unroll
    for (int r = 0; r < 8; ++r) {
      C[(size_t)(row0 + rowoff + r) * NOUT + t * 16 + n0] = acc[t][r];
    }
  }
}

// ---------------------------------------------------------------------------
// alpha_s[i] = dot(h[i,:], a_src), alpha_d[i] = dot(h[i,:], a_dst)
// ---------------------------------------------------------------------------
template <int F>
__global__ void alpha_kernel(const float* __restrict__ h,
                             const float* __restrict__ a_s,
                             const float* __restrict__ a_d,
                             float* __restrict__ out_s,
                             float* __restrict__ out_d, int N) {
  const int i = blockIdx.x * blockDim.x + threadIdx.x;
  if (i >= N) return;
  const float* __restrict__ hr = h + (size_t)i * F;
  float s = 0.f, d = 0.f;
#pragma unroll
  for (int f = 0; f < F; ++f) {
    const float v = hr[f];
    s = fmaf(v, a_s[f], s);
    d = fmaf(v, a_d[f], d);
  }
  out_s[i] = s;
  out_d[i] = d;
}

// order-preserving float<->uint for atomicMax on floats
__device__ __forceinline__ unsigned int f2ord(float f) {
  unsigned int i = __float_as_uint(f);
  return (i & 0x80000000u) ? ~i : (i | 0x80000000u);
}
__device__ __forceinline__ float ord2f(unsigned int u) {
  unsigned int i = (u & 0x80000000u) ? (u ^ 0x80000000u) : ~u;
  return __uint_as_float(i);
}

// pass 1: e = leakyrelu(as[src]+ad[dst]); atomic segment max over dst
__global__ void edge_max_kernel(const int* __restrict__ ei, int E, int ET,
                                const float* __restrict__ as,
                                const float* __restrict__ ad,
                                float* __restrict__ ebuf,
                                unsigned int* __restrict__ mbuf) {
  const int t = blockIdx.x * blockDim.x + threadIdx.x;
  if (t >= ET) return;
  int src, dst;
  if (t < E) { src = ei[t]; dst = ei[E + t]; } else { src = dst = t - E; }
  float e = as[src] + ad[dst];
  e = (e > 0.f) ? e : NEG_SLOPE * e;
  ebuf[t] = e;
  atomicMax(&mbuf[dst], f2ord(e));
}

// pass 2: e = exp(e - m[dst]); atomic segment sum of e over dst
__global__ void edge_exp_kernel(const int* __restrict__ ei, int E, int ET,
                                const unsigned int* __restrict__ mbuf,
                                float* __restrict__ ebuf,
                                float* __restrict__ denom) {
  const int t = blockIdx.x * blockDim.x + threadIdx.x;
  if (t >= ET) return;
  const int dst = (t < E) ? ei[E + t] : (t - E);
  const float ex = expf(ebuf[t] - ord2f(mbuf[dst]));
  ebuf[t] = ex;
  atomicAdd(&denom[dst], ex);
}

// pass 3: out[dst,f] += (e/denom[dst]) * h[src,f]; thread per (edge, feature)
template <int F, int LOGF>
__global__ void edge_scatter_kernel(const int* __restrict__ ei, int E, long long ET,
                                    const float* __restrict__ ebuf,
                                    const float* __restrict__ denom,
                                    const float* __restrict__ h,
                                    float* __restrict__ out) {
  const long long gid = (long long)blockIdx.x * blockDim.x + threadIdx.x;
  if (gid >= ET * F) return;
  const int t = (int)(gid >> LOGF);
  const int f = (int)(gid & (F - 1));
  int src, dst;
  if (t < E) { src = ei[t]; dst = ei[E + t]; } else { src = dst = t - E; }
  const float alpha = ebuf[t] / denom[dst];
  atomicAdd(&out[(size_t)dst * F + f], alpha * h[(size_t)src * F + f]);
}

// bias add (+ optional relu) in place
template <int F, bool RELU>
__global__ void bias_act_kernel(float* __restrict__ h,
                                const float* __restrict__ b, long long total) {
  const long long i = (long long)blockIdx.x * blockDim.x + threadIdx.x;
  if (i >= total) return;
  const int f = (int)(i & (F - 1));
  float v = h[i] + b[f];
  if (RELU) v = fmaxf(v, 0.f);
  h[i] = v;
}

// ---------------------------------------------------------------------------
extern "C" void kernel_launch(void* const* d_in, const int* in_sizes, int n_in,
                              void* d_out, int out_size, void* d_ws, size_t ws_size,
                              hipStream_t stream) {
  constexpr int FIN = 128, HID = 32, FOUT = 64;
  const int N = in_sizes[0] / FIN;
  const int E = in_sizes[1] / 2;
  const int ET = E + N;  // with self loops

  const float* x     = (const float*)d_in[0];
  const int*   ei    = (const int*)d_in[1];   // [2,E] int32 (JAX x64 off)
  const float* W1    = (const float*)d_in[2];
  const float* a1s   = (const float*)d_in[3];
  const float* a1d   = (const float*)d_in[4];
  const float* b1    = (const float*)d_in[5];
  const float* W2    = (const float*)d_in[6];
  const float* a2s   = (const float*)d_in[7];
  const float* a2d   = (const float*)d_in[8];
  const float* b2    = (const float*)d_in[9];
  float* out = (float*)d_out;

  // workspace layout (256B aligned chunks)
  char* ws = (char*)d_ws;
  size_t off = 0;
  auto alloc = [&](size_t bytes) {
    void* p = ws + off;
    off += (bytes + 255) & ~(size_t)255;
    return p;
  };
  float*        h1    = (float*)alloc((size_t)N * HID * 4);
  float*        as1   = (float*)alloc((size_t)N * 4);
  float*        ad1   = (float*)alloc((size_t)N * 4);
  float*        ebuf  = (float*)alloc((size_t)ET * 4);
  unsigned int* mbuf  = (unsigned int*)alloc((size_t)N * 4);
  float*        denom = (float*)alloc((size_t)N * 4);
  float*        agg1  = (float*)alloc((size_t)N * HID * 4);
  float*        h2    = (float*)alloc((size_t)N * FOUT * 4);
  float*        as2   = (float*)alloc((size_t)N * 4);
  float*        ad2   = (float*)alloc((size_t)N * 4);
  (void)ws_size;

  const int BLK = 256;
  const int wavesPerBlock = BLK / 32;
  const int mtiles = (N + 15) / 16;
  const int gemmGrid = (mtiles + wavesPerBlock - 1) / wavesPerBlock;
  const int nGrid  = (N + BLK - 1) / BLK;
  const int eGrid  = (ET + BLK - 1) / BLK;
  const long long sc1 = (long long)ET * HID;
  const long long sc2 = (long long)ET * FOUT;
  const int sc1Grid = (int)((sc1 + BLK - 1) / BLK);
  const int sc2Grid = (int)((sc2 + BLK - 1) / BLK);
  const long long tot1 = (long long)N * HID;
  const long long tot2 = (long long)N * FOUT;

  // ---- layer 1 ----
  wmma_gemm_f32<FIN, HID><<<gemmGrid, BLK, 0, stream>>>(x, W1, h1, N);
  alpha_kernel<HID><<<nGrid, BLK, 0, stream>>>(h1, a1s, a1d, as1, ad1, N);
  hipMemsetAsync(mbuf, 0, (size_t)N * 4, stream);
  edge_max_kernel<<<eGrid, BLK, 0, stream>>>(ei, E, ET, as1, ad1, ebuf, mbuf);
  hipMemsetAsync(denom, 0, (size_t)N * 4, stream);
  edge_exp_kernel<<<eGrid, BLK, 0, stream>>>(ei, E, ET, mbuf, ebuf, denom);
  hipMemsetAsync(agg1, 0, (size_t)N * HID * 4, stream);
  edge_scatter_kernel<HID, 5><<<sc1Grid, BLK, 0, stream>>>(ei, E, (long long)ET,
                                                           ebuf, denom, h1, agg1);
  bias_act_kernel<HID, true><<<(int)((tot1 + BLK - 1) / BLK), BLK, 0, stream>>>(agg1, b1, tot1);

  // ---- layer 2 ----
  wmma_gemm_f32<HID, FOUT><<<gemmGrid, BLK, 0, stream>>>(agg1, W2, h2, N);
  alpha_kernel<FOUT><<<nGrid, BLK, 0, stream>>>(h2, a2s, a2d, as2, ad2, N);
  hipMemsetAsync(mbuf, 0, (size_t)N * 4, stream);
  edge_max_kernel<<<eGrid, BLK, 0, stream>>>(ei, E, ET, as2, ad2, ebuf, mbuf);
  hipMemsetAsync(denom, 0, (size_t)N * 4, stream);
  edge_exp_kernel<<<eGrid, BLK, 0, stream>>>(ei, E, ET, mbuf, ebuf, denom);
  hipMemsetAsync(out, 0, (size_t)N * FOUT * 4, stream);
  edge_scatter_kernel<FOUT, 6><<<sc2Grid, BLK, 0, stream>>>(ei, E, (long long)ET,
                                                            ebuf, denom, h2, out);
  bias_act_kernel<FOUT, false><<<(int)((tot2 + BLK - 1) / BLK), BLK, 0, stream>>>(out, b2, tot2);
}